// GSModel_58523224375842
// MI455X (gfx1250) — compile-verified
//
#include <hip/hip_runtime.h>
#include <math.h>

typedef __attribute__((ext_vector_type(2))) float v2f;
typedef __attribute__((ext_vector_type(8))) float v8f;

#define DFEAT 256

// ---------------------------------------------------------------------------
// aggr = 2*x   (GIN eps=1 -> (1+eps)*x term), then scatter-add of x[src] later
// ---------------------------------------------------------------------------
__global__ void k_scale2(const float* __restrict__ x, float* __restrict__ out, int n) {
  int i = blockIdx.x * blockDim.x + threadIdx.x;
  if (i < n) out[i] = 2.0f * x[i];
}

// one wave32 per edge, 8 feature chunks of 32
__global__ void k_scatter(const float* __restrict__ x, const int* __restrict__ ei,
                          float* __restrict__ aggr, int E) {
  int wave = (blockIdx.x * blockDim.x + threadIdx.x) >> 5;
  int lane = threadIdx.x & 31;
  if (wave >= E) return;
  int src = ei[wave];
  int dst = ei[E + wave];
  const float* xs = x + (size_t)src * DFEAT;
  float* ad = aggr + (size_t)dst * DFEAT;
  __builtin_prefetch(xs + lane, 0, 1);  // global_prefetch_b8 on the gather stream
#pragma unroll
  for (int f = 0; f < DFEAT; f += 32)
    atomicAdd(ad + f + lane, xs[f + lane]);
}

// ---------------------------------------------------------------------------
// 16x16 tiled transpose of a DxD matrix (batched via blockIdx.y, stride D*D).
// Lets every GEMM use the contiguous A @ B^T fragment path (b64 loads only).
// ---------------------------------------------------------------------------
__global__ void k_transpose(const float* __restrict__ in, float* __restrict__ out) {
  __shared__ float tile[16][17];
  const float* ib = in + (size_t)blockIdx.y * DFEAT * DFEAT;
  float* ob = out + (size_t)blockIdx.y * DFEAT * DFEAT;
  int bx = blockIdx.x & 15, by = blockIdx.x >> 4;
  int tx = threadIdx.x & 15, ty = threadIdx.x >> 4;
  tile[ty][tx] = ib[(size_t)(by * 16 + ty) * DFEAT + bx * 16 + tx];
  __syncthreads();
  ob[(size_t)(bx * 16 + ty) * DFEAT + by * 16 + tx] = tile[tx][ty];
}

// ---------------------------------------------------------------------------
// fp32 WMMA GEMM, register-blocked 32x32 tile per wave32:
//   C[M x Nc] = A[M x K] @ Bt^T   (Bt stored [Nc x K] row-major)
// 4 accumulators (2 A frags x 2 B frags) -> 4 v_wmma_f32_16x16x4_f32 per
// 4 b64 loads, each fragment feeding two WMMAs.
// A frag (16x4 f32): lane L<16 -> M=L, v0=K0,v1=K1; L>=16 -> M=L-16, v0=K2,v1=K3
// B frag (4x16 f32): same pattern with N instead of M.
// C/D (16x16 f32, 8 VGPRs): lanes 0-15 rows 0..7, lanes 16-31 rows 8..15.
// ---------------------------------------------------------------------------
template <bool RELU, bool BIAS>
__global__ void k_gemm32(const float* __restrict__ A, const float* __restrict__ Bt,
                         const float* __restrict__ bias, float* __restrict__ C,
                         int K, int Nc, int ntilesN,
                         long strideA, long strideB, long strideC) {
  const float* Ab = A + (size_t)blockIdx.y * strideA;
  const float* Bb = Bt + (size_t)blockIdx.y * strideB;
  float* Cb = C + (size_t)blockIdx.y * strideC;

  int wave = (blockIdx.x * blockDim.x + threadIdx.x) >> 5;
  int lane = threadIdx.x & 31;
  int mt = wave / ntilesN;
  int nt = wave - mt * ntilesN;
  int r15 = lane & 15;
  int half = lane >> 4;

  const float* Ap0 = Ab + (size_t)(mt * 32 + r15) * K + half * 2;
  const float* Ap1 = Ap0 + (size_t)16 * K;
  const float* Bp0 = Bb + (size_t)(nt * 32 + r15) * K + half * 2;
  const float* Bp1 = Bp0 + (size_t)16 * K;

  v8f acc00 = {}, acc01 = {}, acc10 = {}, acc11 = {};

#pragma unroll 4
  for (int k = 0; k < K; k += 4) {
    v2f a0 = *(const v2f*)(Ap0 + k);
    v2f a1 = *(const v2f*)(Ap1 + k);
    v2f b0 = *(const v2f*)(Bp0 + k);
    v2f b1 = *(const v2f*)(Bp1 + k);
#if defined(__HIP_DEVICE_COMPILE__)
    acc00 = __builtin_amdgcn_wmma_f32_16x16x4_f32(false, a0, false, b0, (short)0,
                                                  acc00, false, false);
    acc01 = __builtin_amdgcn_wmma_f32_16x16x4_f32(false, a0, false, b1, (short)0,
                                                  acc01, false, false);
    acc10 = __builtin_amdgcn_wmma_f32_16x16x4_f32(false, a1, false, b0, (short)0,
                                                  acc10, false, false);
    acc11 = __builtin_amdgcn_wmma_f32_16x16x4_f32(false, a1, false, b1, (short)0,
                                                  acc11, false, false);
#else
    (void)a0; (void)a1; (void)b0; (void)b1;
#endif
  }

  int c0 = nt * 32 + r15;
  float bv0 = BIAS ? bias[c0] : 0.0f;
  float bv1 = BIAS ? bias[c0 + 16] : 0.0f;
  int mbase = mt * 32 + half * 8;
  float* Cp0 = Cb + (size_t)mbase * Nc + c0;        // rows mbase..mbase+7
  float* Cp1 = Cb + (size_t)(mbase + 16) * Nc + c0; // rows +16
#pragma unroll
  for (int r = 0; r < 8; ++r) {
    float v00 = acc00[r] + bv0, v01 = acc01[r] + bv1;
    float v10 = acc10[r] + bv0, v11 = acc11[r] + bv1;
    if (RELU) {
      v00 = fmaxf(v00, 0.0f); v01 = fmaxf(v01, 0.0f);
      v10 = fmaxf(v10, 0.0f); v11 = fmaxf(v11, 0.0f);
    }
    Cp0[(size_t)r * Nc] = v00;
    Cp0[(size_t)r * Nc + 16] = v01;
    Cp1[(size_t)r * Nc] = v10;
    Cp1[(size_t)r * Nc + 16] = v11;
  }
}

// ---------------------------------------------------------------------------
// BatchNorm over nodes: per-feature sum & sumsq (block = 128 rows)
// ---------------------------------------------------------------------------
__global__ void k_bnstats(const float* __restrict__ h, float* __restrict__ sums) {
  int d = threadIdx.x;
  size_t row0 = (size_t)blockIdx.x * 128;
  float s = 0.0f, s2 = 0.0f;
  for (int r = 0; r < 128; ++r) {
    float v = h[(row0 + r) * DFEAT + d];
    s += v;
    s2 += v * v;
  }
  atomicAdd(&sums[d], s);
  atomicAdd(&sums[DFEAT + d], s2);
}

__global__ void k_bnapply(const float* __restrict__ h, const float* __restrict__ sums,
                          const float* __restrict__ gamma, const float* __restrict__ beta,
                          float* __restrict__ out, int n) {
  int i = blockIdx.x * blockDim.x + threadIdx.x;
  if (i >= n) return;
  int d = i & (DFEAT - 1);
  const float invN = 1.0f / 32768.0f;
  float mu = sums[d] * invN;
  float var = sums[DFEAT + d] * invN - mu * mu;
  float v = (h[i] - mu) * rsqrtf(var + 1e-5f);
  v = gamma[d] * v + beta[d];
  out[i] = fmaxf(v, 0.0f);
}

// ---------------------------------------------------------------------------
// Gumbel noise + temperature:  la = (scores + g)/TAU, TAU = 0.1
// ---------------------------------------------------------------------------
__global__ void k_gumbel(float* __restrict__ la, const float* __restrict__ u, int n) {
  int i = blockIdx.x * blockDim.x + threadIdx.x;
  if (i >= n) return;
  float g = -logf(-logf(u[i] + 1e-20f) + 1e-20f);
  la[i] = (la[i] + g) * 10.0f;
}

// ---------------------------------------------------------------------------
// Sinkhorn: subtract logsumexp along rows / cols of [B,S,S]
// ---------------------------------------------------------------------------
__global__ void k_row_lse(float* __restrict__ la) {
  __shared__ float red[256];
  int b = blockIdx.x >> 8, s = blockIdx.x & 255;
  float* row = la + ((size_t)b * 256 + s) * 256;
  int t = threadIdx.x;
  float v = row[t];
  red[t] = v;
  __syncthreads();
  for (int o = 128; o > 0; o >>= 1) {
    if (t < o) red[t] = fmaxf(red[t], red[t + o]);
    __syncthreads();
  }
  float m = red[0];
  __syncthreads();
  red[t] = expf(v - m);
  __syncthreads();
  for (int o = 128; o > 0; o >>= 1) {
    if (t < o) red[t] += red[t + o];
    __syncthreads();
  }
  row[t] = v - (m + logf(red[0]));
}

__global__ void k_col_lse(float* __restrict__ la) {
  __shared__ float red[256];
  int b = blockIdx.x >> 8, c = blockIdx.x & 255;
  float* base = la + (size_t)b * 65536 + c;
  int t = threadIdx.x;
  float v = base[(size_t)t * 256];
  red[t] = v;
  __syncthreads();
  for (int o = 128; o > 0; o >>= 1) {
    if (t < o) red[t] = fmaxf(red[t], red[t + o]);
    __syncthreads();
  }
  float m = red[0];
  __syncthreads();
  red[t] = expf(v - m);
  __syncthreads();
  for (int o = 128; o > 0; o >>= 1) {
    if (t < o) red[t] += red[t + o];
    __syncthreads();
  }
  base[(size_t)t * 256] = v - (m + logf(red[0]));
}

// colsum(P)[b,t] = sum_s exp(la[b,s,t])
__global__ void k_colsumP(const float* __restrict__ la, float* __restrict__ col) {
  __shared__ float red[256];
  int b = blockIdx.x >> 8, c = blockIdx.x & 255;
  const float* base = la + (size_t)b * 65536 + c;
  int t = threadIdx.x;
  red[t] = expf(base[(size_t)t * 256]);
  __syncthreads();
  for (int o = 128; o > 0; o >>= 1) {
    if (t < o) red[t] += red[t + o];
    __syncthreads();
  }
  if (t == 0) col[blockIdx.x] = red[0];
}

// rowsum(g2)[node] = sum_d g2[node,d]
__global__ void k_rowsum(const float* __restrict__ g2, float* __restrict__ row) {
  __shared__ float red[256];
  int t = threadIdx.x;
  red[t] = g2[(size_t)blockIdx.x * DFEAT + t];
  __syncthreads();
  for (int o = 128; o > 0; o >>= 1) {
    if (t < o) red[t] += red[t + o];
    __syncthreads();
  }
  if (t == 0) row[blockIdx.x] = red[0];
}

// p[b] = sum(g1[b]) - dot(colsumP[b], rowsum_g2[b])
__global__ void k_pvec(const float* __restrict__ g1, const float* __restrict__ col,
                       const float* __restrict__ row, float* __restrict__ p) {
  __shared__ float red[256];
  int b = blockIdx.x, t = threadIdx.x;
  const float* g = g1 + (size_t)b * 65536;
  float acc = 0.0f;
  for (int k = 0; k < 256; ++k) acc += g[(size_t)k * 256 + t];
  acc -= col[b * 256 + t] * row[b * 256 + t];
  red[t] = acc;
  __syncthreads();
  for (int o = 128; o > 0; o >>= 1) {
    if (t < o) red[t] += red[t + o];
    __syncthreads();
  }
  if (t == 0) p[b] = red[0];
}

// normalize p to [0,1] over the batch, out = exp(-p)
__global__ void k_final(const float* __restrict__ p, float* __restrict__ out) {
  __shared__ float mn[128], mx[128];
  int t = threadIdx.x;
  float v = p[t];
  mn[t] = v;
  mx[t] = v;
  __syncthreads();
  for (int o = 64; o > 0; o >>= 1) {
    if (t < o) {
      mn[t] = fminf(mn[t], mn[t + o]);
      mx[t] = fmaxf(mx[t], mx[t + o]);
    }
    __syncthreads();
  }
  float lo = mn[0], hi = mx[0];
  out[t] = expf(-((v - lo) / (hi - lo)));
}

// ---------------------------------------------------------------------------
extern "C" void kernel_launch(void* const* d_in, const int* in_sizes, int n_in,
                              void* d_out, int out_size, void* d_ws, size_t ws_size,
                              hipStream_t stream) {
  const float* f1 = (const float*)d_in[0];
  const float* f2 = (const float*)d_in[1];
  const int* e1 = (const int*)d_in[2];
  const int* e2 = (const int*)d_in[3];
  const float* un = (const float*)d_in[4];
  const float* gw1 = (const float*)d_in[5];
  const float* gb1 = (const float*)d_in[6];
  const float* gw2 = (const float*)d_in[7];
  const float* gb2 = (const float*)d_in[8];
  const float* gga = (const float*)d_in[9];
  const float* gbe = (const float*)d_in[10];
  const float* fc1w = (const float*)d_in[11];
  const float* fc1b = (const float*)d_in[12];
  const float* fc2w = (const float*)d_in[13];
  const float* fc2b = (const float*)d_in[14];
  const float* Wm = (const float*)d_in[15];
  float* out = (float*)d_out;

  const int Nn = 32768, Bb = 128, S = 256, Dd = 256, Ll = 3;
  const int E = in_sizes[2] / 2;
  const size_t NS = (size_t)Nn * Dd;  // 8388608 floats
  const size_t DD = (size_t)Dd * Dd;  // 65536 floats

  float* ws = (float*)d_ws;
  float* X1 = ws;                 // g1 (GIN output side 1)
  float* X2 = ws + NS;            // g2
  float* T1 = ws + 2 * NS;        // temp / later aliased as log_alpha
  float* T2 = ws + 3 * NS;        // temp
  float* M2 = ws + 4 * NS;        // transform(g2)
  float* MW = ws + 5 * NS;        // transform(g1) @ W
  float* LA = T1;                 // log_alpha [B,S,S] (= NS floats)
  float* COL = ws + 6 * NS;       // colsum(P)  [B*S]
  float* ROW = COL + (size_t)Bb * S;  // rowsum(g2) [B*S]
  float* SUM = ROW + (size_t)Bb * S;  // BN sum/sumsq [2*D]
  float* PV = SUM + 2 * Dd;           // p [B]
  // transposed weights ([N x K] layouts so all GEMMs use contiguous b64 loads)
  float* Tgw1 = PV + Bb;          // 3 x DD
  float* Tgw2 = Tgw1 + 3 * DD;    // 3 x DD
  float* Tfc1 = Tgw2 + 3 * DD;
  float* Tfc2 = Tfc1 + DD;
  float* TWm = Tfc2 + DD;

  dim3 blk(256);
  const int nsBlocks = (int)(NS / 256);
  const int gemmBlocksFull = (Nn / 32) * (Dd / 32) / 8;  // 1024 blocks (8 waves/blk)
  const int ntN = Dd / 32;                               // 8 col-tiles

  // ---- transpose all weight matrices once ----
  k_transpose<<<dim3(256, 3), blk, 0, stream>>>(gw1, Tgw1);
  k_transpose<<<dim3(256, 3), blk, 0, stream>>>(gw2, Tgw2);
  k_transpose<<<dim3(256, 1), blk, 0, stream>>>(fc1w, Tfc1);
  k_transpose<<<dim3(256, 1), blk, 0, stream>>>(fc2w, Tfc2);
  k_transpose<<<dim3(256, 1), blk, 0, stream>>>(Wm, TWm);

  // ---- GIN stacks (both sides) ----
  for (int side = 0; side < 2; ++side) {
    const float* feat = side ? f2 : f1;
    const int* ei = side ? e2 : e1;
    float* X = side ? X2 : X1;
    for (int l = 0; l < Ll; ++l) {
      const float* xc = (l == 0) ? feat : X;
      k_scale2<<<nsBlocks, blk, 0, stream>>>(xc, T2, (int)NS);
      k_scatter<<<E / 8, blk, 0, stream>>>(xc, ei, T2, E);
      k_gemm32<true, true><<<dim3(gemmBlocksFull, 1), blk, 0, stream>>>(
          T2, Tgw1 + l * DD, gb1 + l * Dd, T1, Dd, Dd, ntN, 0, 0, 0);
      k_gemm32<false, true><<<dim3(gemmBlocksFull, 1), blk, 0, stream>>>(
          T1, Tgw2 + l * DD, gb2 + l * Dd, T2, Dd, Dd, ntN, 0, 0, 0);
      hipMemsetAsync(SUM, 0, 2 * Dd * sizeof(float), stream);
      k_bnstats<<<Nn / 128, blk, 0, stream>>>(T2, SUM);
      k_bnapply<<<nsBlocks, blk, 0, stream>>>(T2, SUM, gga + l * Dd, gbe + l * Dd, X,
                                              (int)NS);
    }
  }

  // ---- fc transforms + bilinear left side ----
  k_gemm32<true, true><<<dim3(gemmBlocksFull, 1), blk, 0, stream>>>(
      X1, Tfc1, fc1b, T1, Dd, Dd, ntN, 0, 0, 0);
  k_gemm32<false, true><<<dim3(gemmBlocksFull, 1), blk, 0, stream>>>(
      T1, Tfc2, fc2b, T2, Dd, Dd, ntN, 0, 0, 0);
  k_gemm32<false, false><<<dim3(gemmBlocksFull, 1), blk, 0, stream>>>(
      T2, TWm, nullptr, MW, Dd, Dd, ntN, 0, 0, 0);
  k_gemm32<true, true><<<dim3(gemmBlocksFull, 1), blk, 0, stream>>>(
      X2, Tfc1, fc1b, T1, Dd, Dd, ntN, 0, 0, 0);
  k_gemm32<false, true><<<dim3(gemmBlocksFull, 1), blk, 0, stream>>>(
      T1, Tfc2, fc2b, M2, Dd, Dd, ntN, 0, 0, 0);

  // ---- scores[b] = (m1@W)[b] @ m2[b]^T -> LA  (m2 is already the "Bt") ----
  k_gemm32<false, false><<<dim3((S / 32) * (S / 32) / 8, Bb), blk, 0, stream>>>(
      MW, M2, nullptr, LA, Dd, S, S / 32, (long)S * Dd, (long)S * Dd, (long)S * S);

  // ---- Gumbel + Sinkhorn ----
  k_gumbel<<<(Bb * S * S) / 256, blk, 0, stream>>>(LA, un, Bb * S * S);
  for (int it = 0; it < 20; ++it) {
    k_row_lse<<<Bb * S, blk, 0, stream>>>(LA);
    k_col_lse<<<Bb * S, blk, 0, stream>>>(LA);
  }

  // ---- p[b] = sum(g1) - colsum(P).rowsum(g2); out = exp(-normalized p) ----
  k_colsumP<<<Bb * S, blk, 0, stream>>>(LA, COL);
  k_rowsum<<<Nn, blk, 0, stream>>>(X2, ROW);
  k_pvec<<<Bb, blk, 0, stream>>>(X1, COL, ROW, PV);
  k_final<<<1, 128, 0, stream>>>(PV, out);
}